// CrissCrossAttention_635655160209
// MI455X (gfx1250) — compile-verified
//
#include <hip/hip_runtime.h>
#include <math.h>

#define NB 16
#define NC 64
#define NQ 8
#define NH 128
#define NW 128
#define NHW (NH*NW)

typedef __attribute__((ext_vector_type(2))) float v2f;
typedef __attribute__((ext_vector_type(8))) float v8f;

static __device__ __forceinline__ v8f v8f_zero() {
  v8f z;
#pragma unroll
  for (int i = 0; i < 8; ++i) z[i] = 0.f;
  return z;
}

static __device__ __forceinline__ v8f wmma4(v2f a, v2f b, v8f c) {
  // V_WMMA_F32_16X16X4_F32 : D = A(16x4) * B(4x16) + C(16x16)
  return __builtin_amdgcn_wmma_f32_16x16x4_f32(false, a, false, b, (short)0, c,
                                               false, false);
}

// ---- CDNA5 async global->LDS DMA (ASYNCcnt-tracked, no VGPR round trip) ----
// Generic pointers to __shared__ carry the LDS byte offset in addr[31:0]
// (aperture check discards the upper bits), so truncation gives the DS offset.
static __device__ __forceinline__ void async_cp_b32(void* lds, const void* g) {
  unsigned l = (unsigned)(unsigned long long)(uintptr_t)lds;
  asm volatile("global_load_async_to_lds_b32 %0, %1, off"
               :: "v"(l), "v"(g) : "memory");
}
static __device__ __forceinline__ void async_cp_b128(void* lds, const void* g) {
  unsigned l = (unsigned)(unsigned long long)(uintptr_t)lds;
  asm volatile("global_load_async_to_lds_b128 %0, %1, off"
               :: "v"(l), "v"(g) : "memory");
}
static __device__ __forceinline__ void wait_async() {
  asm volatile("s_wait_asynccnt 0x0" ::: "memory");
}

static __device__ __forceinline__ float redmax16(float v) {
  v = fmaxf(v, __shfl_xor(v, 8, 32));
  v = fmaxf(v, __shfl_xor(v, 4, 32));
  v = fmaxf(v, __shfl_xor(v, 2, 32));
  v = fmaxf(v, __shfl_xor(v, 1, 32));
  return v;
}
static __device__ __forceinline__ float redsum16(float v) {
  v += __shfl_xor(v, 8, 32);
  v += __shfl_xor(v, 4, 32);
  v += __shfl_xor(v, 2, 32);
  v += __shfl_xor(v, 1, 32);
  return v;
}

// Energy tile: D(16x16) = Q^T K over K=8 channels, Qs/Ks stored [c][pos], row
// stride NH. A[m,k] = Qs[k][m0+m], B[k,n] = Ks[k][n0+n].
static __device__ __forceinline__ v8f energy_tile(const float* Qs, const float* Ks,
                                                  int m0, int n0, int hf, int l16) {
  v8f acc = v8f_zero();
#pragma unroll
  for (int k0 = 0; k0 < NQ; k0 += 4) {
    int kk = k0 + 2*hf;
    v2f a, b;
    a.x = Qs[kk*NH + m0 + l16];
    a.y = Qs[(kk+1)*NH + m0 + l16];
    b.x = Ks[kk*NH + n0 + l16];
    b.y = Ks[(kk+1)*NH + n0 + l16];
    acc = wmma4(a, b, acc);
  }
  return acc;
}

// Online softmax stats update across 16-lane row groups.
static __device__ __forceinline__ void online_update(float e, float& M, float& S) {
  float tmax = redmax16(e);
  float p = __expf(e - tmax);
  float tsum = redsum16(p);
  float Mn = fmaxf(M, tmax);
  S = S*__expf(M - Mn) + tsum*__expf(tmax - Mn);
  M = Mn;
}

// ---------------- Kernel 1: fused QKV projection GEMM ----------------
// qkv[m, p] = sum_c Wqkv[m,c] * x[b,c,p] + bias[m];  m: 0..7 q, 8..15 k, 16..79 v
__global__ __launch_bounds__(256) void k_proj(
    const float* __restrict__ x,
    const float* __restrict__ Wq, const float* __restrict__ bq,
    const float* __restrict__ Wk, const float* __restrict__ bk,
    const float* __restrict__ Wv, const float* __restrict__ bv,
    float* __restrict__ qb, float* __restrict__ kb, float* __restrict__ vb) {
  __shared__ float Wl[80*65];   // stride 65: conflict-free A-frag loads
  __shared__ float bl[80];
  int tid = threadIdx.x;
  for (int i = tid; i < 80*64; i += 256) {
    int m = i >> 6, c = i & 63;
    float wv;
    if (m < 8)       wv = Wq[m*NC + c];
    else if (m < 16) wv = Wk[(m-8)*NC + c];
    else             wv = Wv[(m-16)*NC + c];
    Wl[m*65 + c] = wv;
  }
  if (tid < 80) bl[tid] = (tid < 8) ? bq[tid] : ((tid < 16) ? bk[tid-8] : bv[tid-16]);
  __syncthreads();

  int b = blockIdx.x >> 7;
  int tile = blockIdx.x & 127;
  int lane = tid & 31, wid = tid >> 5;
  int hf = lane >> 4, l16 = lane & 15;
  int p0 = tile*128 + wid*16;            // this wave's 16-pixel N-tile
  const float* xb = x + (size_t)b*NC*NHW;

  v8f acc[5];
#pragma unroll
  for (int t = 0; t < 5; ++t) acc[t] = v8f_zero();
  for (int c0 = 0; c0 < NC; c0 += 4) {
    int kk = c0 + 2*hf;
    v2f bf;                               // B[k,n] = x[b, k, p0+n]
    bf.x = xb[(size_t)kk*NHW + p0 + l16];
    bf.y = xb[(size_t)(kk+1)*NHW + p0 + l16];
#pragma unroll
    for (int t = 0; t < 5; ++t) {
      v2f af;                             // A[m,k] = Wl[m0+m][k]
      af.x = Wl[(t*16 + l16)*65 + kk];
      af.y = Wl[(t*16 + l16)*65 + kk + 1];
      acc[t] = wmma4(af, bf, acc[t]);
    }
  }
  int p = p0 + l16;
#pragma unroll
  for (int t = 0; t < 5; ++t) {
#pragma unroll
    for (int r = 0; r < 8; ++r) {
      int m = t*16 + r + hf*8;
      float val = acc[t][r] + bl[m];
      if (m < 8)        qb[((size_t)b*NQ + m)*NHW + p] = val;
      else if (m < 16)  kb[((size_t)b*NQ + (m-8))*NHW + p] = val;
      else              vb[((size_t)b*NC + (m-16))*NHW + p] = val;
    }
  }
}

// ---------------- Kernel 2: column-attention softmax stats ----------------
// Per (b,w): EH[h,g] = q[:,h,w].k[:,g,w], diag masked; write (mH,sH)[b,h,w].
__global__ __launch_bounds__(256) void k_colstats(
    const float* __restrict__ qb, const float* __restrict__ kb,
    float* __restrict__ mH, float* __restrict__ sH) {
  __shared__ float Qs[NQ*NH];
  __shared__ float Ks[NQ*NH];
  int b = blockIdx.x >> 7, w = blockIdx.x & 127;
  int tid = threadIdx.x;
  for (int i = tid; i < NQ*NH; i += 256) {
    int c = i >> 7, h = i & 127;
    size_t gidx = ((size_t)(b*NQ + c)*NH + h)*NW + w;
    Qs[c*NH + h] = qb[gidx];
    Ks[c*NH + h] = kb[gidx];
  }
  __syncthreads();
  int lane = tid & 31, wid = tid >> 5;
  int hf = lane >> 4, l16 = lane & 15;
  int h0 = wid*16;
  float M[8], S[8];
#pragma unroll
  for (int r = 0; r < 8; ++r) { M[r] = -INFINITY; S[r] = 0.f; }
  for (int g0 = 0; g0 < NH; g0 += 16) {
    v8f acc = energy_tile(Qs, Ks, h0, g0, hf, l16);
    int g = g0 + l16;
#pragma unroll
    for (int r = 0; r < 8; ++r) {
      int h = h0 + r + hf*8;
      float e = (h == g) ? -INFINITY : acc[r];
      online_update(e, M[r], S[r]);
    }
  }
  if (l16 == 0) {
#pragma unroll
    for (int r = 0; r < 8; ++r) {
      int h = h0 + r + hf*8;
      size_t idx = ((size_t)b*NH + h)*NW + w;
      mH[idx] = M[r];
      sH[idx] = S[r];
    }
  }
}

// ---------------- Kernel 3: row stats + merge into combined (M,S) ----------------
__global__ __launch_bounds__(256) void k_rowstats(
    const float* __restrict__ qb, const float* __restrict__ kb,
    const float* __restrict__ mH, const float* __restrict__ sH,
    float* __restrict__ Mc, float* __restrict__ Sc) {
  __shared__ float Qs[NQ*NW];
  __shared__ float Ks[NQ*NW];
  int b = blockIdx.x >> 7, h = blockIdx.x & 127;
  int tid = threadIdx.x;
  for (int i = tid; i < NQ*NW; i += 256) {
    int c = i >> 7, wv = i & 127;
    size_t gidx = ((size_t)(b*NQ + c)*NH + h)*NW + wv;
    Qs[c*NW + wv] = qb[gidx];
    Ks[c*NW + wv] = kb[gidx];
  }
  __syncthreads();
  int lane = tid & 31, wid = tid >> 5;
  int hf = lane >> 4, l16 = lane & 15;
  int w0 = wid*16;
  float M[8], S[8];
#pragma unroll
  for (int r = 0; r < 8; ++r) { M[r] = -INFINITY; S[r] = 0.f; }
  for (int v0 = 0; v0 < NW; v0 += 16) {
    v8f acc = energy_tile(Qs, Ks, w0, v0, hf, l16);
#pragma unroll
    for (int r = 0; r < 8; ++r) online_update(acc[r], M[r], S[r]);
  }
  if (l16 == 0) {
#pragma unroll
    for (int r = 0; r < 8; ++r) {
      int w = w0 + r + hf*8;
      size_t idx = ((size_t)b*NH + h)*NW + w;
      float mh = mH[idx], sh = sH[idx];
      float Mn = fmaxf(mh, M[r]);
      float Sn = sh*__expf(mh - Mn) + S[r]*__expf(M[r] - Mn);
      Mc[idx] = Mn;
      Sc[idx] = Sn;
    }
  }
}

// ---------------- Kernel 4: column apply: out = V_col x attH^T ----------------
__global__ __launch_bounds__(256) void k_colapply(
    const float* __restrict__ qb, const float* __restrict__ kb,
    const float* __restrict__ vb, const float* __restrict__ Mc,
    const float* __restrict__ Sc, float* __restrict__ out) {
  __shared__ float Qs[NQ*NH];
  __shared__ float Ks[NQ*NH];
  __shared__ float Vs[NC*(NH+1)];    // [c][g], stride 129
  __shared__ float At[NH*(NH+1)];    // [g][h], stride 129 (attH transposed)
  __shared__ float Ml[NH];
  __shared__ float Sl[NH];
  int b = blockIdx.x >> 7, w = blockIdx.x & 127;
  int tid = threadIdx.x;
  // Issue V-column DMA first: it is only needed after phase 1, so it overlaps
  // the energy-GEMM + softmax work below (ASYNCcnt tracked).
  for (int i = tid; i < NC*NH; i += 256) {
    int c = i >> 7, g = i & 127;
    async_cp_b32(&Vs[c*(NH+1) + g], &vb[((size_t)(b*NC + c)*NH + g)*NW + w]);
  }
  for (int i = tid; i < NQ*NH; i += 256) {
    int c = i >> 7, h = i & 127;
    size_t gidx = ((size_t)(b*NQ + c)*NH + h)*NW + w;
    Qs[c*NH + h] = qb[gidx];
    Ks[c*NH + h] = kb[gidx];
  }
  if (tid < NH) {
    size_t idx = ((size_t)b*NH + tid)*NW + w;
    Ml[tid] = Mc[idx];
    Sl[tid] = 1.0f / Sc[idx];
  }
  __syncthreads();
  int lane = tid & 31, wid = tid >> 5;
  int hf = lane >> 4, l16 = lane & 15;
  int h0 = wid*16;
  for (int g0 = 0; g0 < NH; g0 += 16) {
    v8f acc = energy_tile(Qs, Ks, h0, g0, hf, l16);
    int g = g0 + l16;
#pragma unroll
    for (int r = 0; r < 8; ++r) {
      int h = h0 + r + hf*8;
      float e = (h == g) ? -INFINITY : acc[r];
      At[g*(NH+1) + h] = __expf(e - Ml[h]) * Sl[h];
    }
  }
  wait_async();      // my wave's V DMA done; barrier makes it WG-wide
  __syncthreads();
  // outH[c,h] = sum_g Vs[c][g] * At[g][h]  (M=c tiles, N=h tiles, K=128)
  int mt = wid & 3;
  int nb = (wid >> 2)*4;
  v8f acc2[4];
#pragma unroll
  for (int t = 0; t < 4; ++t) acc2[t] = v8f_zero();
  for (int g0 = 0; g0 < NH; g0 += 4) {
    int kk = g0 + 2*hf;
    v2f af;
    af.x = Vs[(mt*16 + l16)*(NH+1) + kk];
    af.y = Vs[(mt*16 + l16)*(NH+1) + kk + 1];
#pragma unroll
    for (int t = 0; t < 4; ++t) {
      int n0 = (nb + t)*16;
      v2f bf;
      bf.x = At[kk*(NH+1) + n0 + l16];
      bf.y = At[(kk+1)*(NH+1) + n0 + l16];
      acc2[t] = wmma4(af, bf, acc2[t]);
    }
  }
#pragma unroll
  for (int t = 0; t < 4; ++t) {
    int h = (nb + t)*16 + l16;
#pragma unroll
    for (int r = 0; r < 8; ++r) {
      int c = mt*16 + r + hf*8;
      out[((size_t)(b*NC + c)*NH + h)*NW + w] = acc2[t][r];
    }
  }
}

// ---------------- Kernel 5: row apply + fused epilogue ----------------
__global__ __launch_bounds__(256) void k_rowapply(
    const float* __restrict__ qb, const float* __restrict__ kb,
    const float* __restrict__ vb, const float* __restrict__ Mc,
    const float* __restrict__ Sc, const float* __restrict__ x,
    const float* __restrict__ gamma, float* __restrict__ out) {
  __shared__ float Qs[NQ*NW];
  __shared__ float Ks[NQ*NW];
  __shared__ float Vs[NC*(NW+1)];    // [c][v]
  __shared__ float At[NW*(NW+1)];    // [v][w] (attW transposed)
  __shared__ float Ml[NW];
  __shared__ float Sl[NW];
  int b = blockIdx.x >> 7, h = blockIdx.x & 127;
  int tid = threadIdx.x;
  float gm = gamma[0];
  // V-row is contiguous: async 128-bit DMA into LDS, overlapped with phase 1.
  for (int i = tid; i < (NC*NW)/4; i += 256) {
    int e = i*4;
    int c = e >> 7, wv = e & 127;
    async_cp_b128(&Vs[c*(NW+1) + wv],
                  &vb[((size_t)(b*NC + c)*NH + h)*NW + wv]);
  }
  for (int i = tid; i < NQ*NW; i += 256) {
    int c = i >> 7, wv = i & 127;
    size_t gidx = ((size_t)(b*NQ + c)*NH + h)*NW + wv;
    Qs[c*NW + wv] = qb[gidx];
    Ks[c*NW + wv] = kb[gidx];
  }
  if (tid < NW) {
    size_t idx = ((size_t)b*NH + h)*NW + tid;
    Ml[tid] = Mc[idx];
    Sl[tid] = 1.0f / Sc[idx];
  }
  __syncthreads();
  int lane = tid & 31, wid = tid >> 5;
  int hf = lane >> 4, l16 = lane & 15;
  int w0 = wid*16;
  for (int v0 = 0; v0 < NW; v0 += 16) {
    v8f acc = energy_tile(Qs, Ks, w0, v0, hf, l16);
    int vcol = v0 + l16;
#pragma unroll
    for (int r = 0; r < 8; ++r) {
      int wq = w0 + r + hf*8;
      At[vcol*(NW+1) + wq] = __expf(acc[r] - Ml[wq]) * Sl[wq];
    }
  }
  wait_async();
  __syncthreads();
  // outW[c,w] = sum_v Vs[c][v] * At[v][w]
  int mt = wid & 3;
  int nb = (wid >> 2)*4;
  v8f acc2[4];
#pragma unroll
  for (int t = 0; t < 4; ++t) acc2[t] = v8f_zero();
  for (int v0 = 0; v0 < NW; v0 += 4) {
    int kk = v0 + 2*hf;
    v2f af;
    af.x = Vs[(mt*16 + l16)*(NW+1) + kk];
    af.y = Vs[(mt*16 + l16)*(NW+1) + kk + 1];
#pragma unroll
    for (int t = 0; t < 4; ++t) {
      int n0 = (nb + t)*16;
      v2f bf;
      bf.x = At[kk*(NW+1) + n0 + l16];
      bf.y = At[(kk+1)*(NW+1) + n0 + l16];
      acc2[t] = wmma4(af, bf, acc2[t]);
    }
  }
#pragma unroll
  for (int t = 0; t < 4; ++t) {
    int wq = (nb + t)*16 + l16;
#pragma unroll
    for (int r = 0; r < 8; ++r) {
      int c = mt*16 + r + hf*8;
      size_t idx = ((size_t)(b*NC + c)*NH + h)*NW + wq;
      out[idx] = gm*(out[idx] + acc2[t][r]) + x[idx];
    }
  }
}

extern "C" void kernel_launch(void* const* d_in, const int* in_sizes, int n_in,
                              void* d_out, int out_size, void* d_ws, size_t ws_size,
                              hipStream_t stream) {
  const float* x     = (const float*)d_in[0];
  const float* Wq    = (const float*)d_in[1];
  const float* bq    = (const float*)d_in[2];
  const float* Wk    = (const float*)d_in[3];
  const float* bk    = (const float*)d_in[4];
  const float* Wv    = (const float*)d_in[5];
  const float* bv    = (const float*)d_in[6];
  const float* gamma = (const float*)d_in[7];
  float* ws = (float*)d_ws;

  const size_t QSZ = (size_t)NB*NQ*NHW;   // 2,097,152 floats
  const size_t VSZ = (size_t)NB*NC*NHW;   // 16,777,216 floats
  const size_t PSZ = (size_t)NB*NHW;      //   262,144 floats
  float* qb = ws;
  float* kb = qb + QSZ;
  float* vb = kb + QSZ;
  float* mH = vb + VSZ;
  float* sH = mH + PSZ;
  float* Mc = sH + PSZ;
  float* Sc = Mc + PSZ;
  float* out = (float*)d_out;

  dim3 blk(256);
  dim3 grd(NB*128);   // 2048 workgroups for every stage
  k_proj<<<grd, blk, 0, stream>>>(x, Wq, bq, Wk, bk, Wv, bv, qb, kb, vb);
  k_colstats<<<grd, blk, 0, stream>>>(qb, kb, mH, sH);
  k_rowstats<<<grd, blk, 0, stream>>>(qb, kb, mH, sH, Mc, Sc);
  k_colapply<<<grd, blk, 0, stream>>>(qb, kb, vb, Mc, Sc, out);
  k_rowapply<<<grd, blk, 0, stream>>>(qb, kb, vb, Mc, Sc, x, gamma, out);
}